// GraphSAGE_17300128268562
// MI455X (gfx1250) — compile-verified
//
#include <hip/hip_runtime.h>
#include <cstdint>

typedef __attribute__((ext_vector_type(2))) float v2f;
typedef __attribute__((ext_vector_type(8))) float v8f;
typedef unsigned int u32x4 __attribute__((ext_vector_type(4)));
typedef int          i32x4 __attribute__((ext_vector_type(4)));
typedef int          i32x8 __attribute__((ext_vector_type(8)));

#define N_FEAT    128
#define OUT_FEAT  64
#define TILE_ROWS 32
#define TILE_ELEMS (TILE_ROWS * N_FEAT)   // 4096 f32 = 16 KB, contiguous
#define TILES_PER_BLOCK 4

#if defined(__AMDGCN__) && __has_builtin(__builtin_amdgcn_tensor_load_to_lds) && \
    __has_builtin(__builtin_amdgcn_s_wait_tensorcnt)
#define GSAGE_TDM 1
#endif

#ifdef GSAGE_TDM
// 1-D TDM copy: n_valid elements (f32) from gsrc -> LDS; reads past n_valid
// (up to TILE_ELEMS) return zero per TDM OOB semantics.
__device__ inline void gsage_tdm_load_1d(const float* gsrc, float* ldst,
                                         unsigned int n_valid) {
    unsigned long long ga = (unsigned long long)(uintptr_t)gsrc;
    unsigned int lds_off  = (unsigned int)(unsigned long long)(uintptr_t)ldst;
    u32x4 g0;
    g0[0] = 1u;                                   // count=1, user descriptor
    g0[1] = lds_off;                              // LDS byte address
    g0[2] = (unsigned int)(ga & 0xFFFFFFFFu);     // global_addr[31:0]
    g0[3] = (unsigned int)((ga >> 32) & 0x01FFFFFFu) | (2u << 30); // type=2
    i32x8 g1;
    g1[0] = 0x00020000;                           // data_size=4B; no multicast
    g1[1] = (int)((n_valid & 0xFFFFu) << 16);     // tensor_dim0[15:0]
    g1[2] = (int)((n_valid >> 16) & 0xFFFFu) | (1 << 16); // dim0 hi | tensor_dim1=1
    g1[3] = (int)((TILE_ELEMS & 0xFFFFu) << 16);  // tile_dim0 = 4096 (1 row)
    g1[4] = 0;                                    // tile_dim1=0 (1-D), tile_dim2=0
    g1[5] = TILE_ELEMS;                           // tensor_dim0_stride (unused)
    g1[6] = 0;
    g1[7] = 0;
    i32x4 z4 = {0, 0, 0, 0};
#if __clang_major__ >= 23
    i32x8 z8 = {0, 0, 0, 0, 0, 0, 0, 0};
    __builtin_amdgcn_tensor_load_to_lds(g0, g1, z4, z4, z8, 0);
#else
    __builtin_amdgcn_tensor_load_to_lds(g0, g1, z4, z4, 0);
#endif
}
#endif

// ---------------------------------------------------------------- zero fill
__global__ void gsage_zero_f32(float* __restrict__ p, long long n) {
    long long i = (long long)blockIdx.x * blockDim.x + threadIdx.x;
    long long stride = (long long)gridDim.x * blockDim.x;
    for (; i < n; i += stride) p[i] = 0.0f;
}

// ---------------------------------------------------------------- degree
__global__ void gsage_degree(const long long* __restrict__ dst,
                             float* __restrict__ deg, int n_edges) {
    int e = blockIdx.x * blockDim.x + threadIdx.x;
    if (e < n_edges) atomicAdd(&deg[(int)dst[e]], 1.0f);
}

__global__ void gsage_invdeg(float* __restrict__ deg, int n) {
    int i = blockIdx.x * blockDim.x + threadIdx.x;
    if (i < n) deg[i] = 1.0f / fmaxf(deg[i], 1.0f);
}

// ------------------------------------------------- scatter-add (one wave/edge)
__global__ void gsage_scatter_add(const float* __restrict__ H,
                                  const long long* __restrict__ src,
                                  const long long* __restrict__ dst,
                                  float* __restrict__ Agg, int n_edges) {
    long long gid = (long long)blockIdx.x * blockDim.x + threadIdx.x;
    int e = (int)(gid >> 5);
    if (e >= n_edges) return;
    int c = ((int)gid & 31) << 2;
    long long s = src[e];
    long long d = dst[e];
    float4 v = *(const float4*)(H + s * N_FEAT + c);    // coalesced 512B/wave
    float* out = Agg + d * N_FEAT + c;                  // L2-resident atomics
    atomicAdd(out + 0, v.x);
    atomicAdd(out + 1, v.y);
    atomicAdd(out + 2, v.z);
    atomicAdd(out + 3, v.w);
}

// ---------------------------------------------- fused SAGE layer (WMMA f32)
// Out[i,:] = relu( invdeg[i]*(Agg[i,:] @ Wl) + H[i,:] @ Wr + b )
// 8 waves; TILES_PER_BLOCK tiles of 32 rows, double-buffered TDM staging.
// Each wave: 16-col slice, two stacked 16x16 blocks sharing B fragments.
__global__ __launch_bounds__(256)
void gsage_sage_layer(const float* __restrict__ Agg,
                      const float* __restrict__ H,
                      const float* __restrict__ invdeg,
                      const float* __restrict__ Wl,   // [128,128] K-major rows
                      const float* __restrict__ Wr,   // [128,128]
                      const float* __restrict__ bias, // [128]
                      float* __restrict__ Out,
                      int n_rows, int n_tiles) {
    __shared__ float ldsA[2][TILE_ELEMS];
    __shared__ float ldsH[2][TILE_ELEMS];

    const int tile0 = blockIdx.x * TILES_PER_BLOCK;
    int nt = n_tiles - tile0;
    if (nt <= 0) return;
    if (nt > TILES_PER_BLOCK) nt = TILES_PER_BLOCK;

    const int tid   = threadIdx.x;
    const int wave  = tid >> 5;
    const int lane  = tid & 31;
    const int n_off = wave * 16;
    const int l15   = lane & 15;
    const int hi    = lane >> 4;
    const float bval = bias[n_off + l15];

    const long long total = (long long)n_rows * N_FEAT;

#ifdef GSAGE_TDM
    // prologue: issue tile 0 into buffer 0 (wave 0 only; EXEC ignored by TDM)
    if (wave == 0) {
        long long base = (long long)tile0 * TILE_ELEMS;
        unsigned int nv = (unsigned int)((total - base) < TILE_ELEMS
                                             ? (total - base) : TILE_ELEMS);
        gsage_tdm_load_1d(Agg + base, &ldsA[0][0], nv);
        gsage_tdm_load_1d(H   + base, &ldsH[0][0], nv);
    }
#endif

    for (int t = 0; t < nt; ++t) {
        const int buf = t & 1;
#ifdef GSAGE_TDM
        if (wave == 0) {
            if (t + 1 < nt) {   // prefetch next tile into the other buffer
                long long base = (long long)(tile0 + t + 1) * TILE_ELEMS;
                unsigned int nv = (unsigned int)((total - base) < TILE_ELEMS
                                                     ? (total - base) : TILE_ELEMS);
                gsage_tdm_load_1d(Agg + base, &ldsA[buf ^ 1][0], nv);
                gsage_tdm_load_1d(H   + base, &ldsH[buf ^ 1][0], nv);
                __builtin_amdgcn_s_wait_tensorcnt(2);   // current tile done
            } else {
                __builtin_amdgcn_s_wait_tensorcnt(0);
            }
        }
        __syncthreads();
#else
        // fallback: synchronous cooperative staging (pure copy)
        {
            long long base = (long long)(tile0 + t) * TILE_ELEMS;
            for (int i = tid; i < TILE_ELEMS / 4; i += 256) {
                long long g = base + i * 4;
                float4 a = {0, 0, 0, 0}, h = {0, 0, 0, 0};
                if (g + 3 < total) {
                    a = *(const float4*)(Agg + g);
                    h = *(const float4*)(H + g);
                }
                *(float4*)(&ldsA[buf][i * 4]) = a;
                *(float4*)(&ldsH[buf][i * 4]) = h;
            }
            __syncthreads();
        }
#endif
        // ---- WMMA mainloop: separate agg / self accumulators ----
        v8f cA0, cA1, cH0, cH1;
#pragma unroll
        for (int j = 0; j < 8; ++j) {
            cA0[j] = 0.0f; cA1[j] = 0.0f; cH0[j] = bval; cH1[j] = bval;
        }
        const float* lA = &ldsA[buf][0];
        const float* lH = &ldsH[buf][0];
#pragma unroll 4
        for (int kk = 0; kk < N_FEAT / 4; ++kk) {
            int kb = 4 * kk + (hi << 1);
            v2f bL, bR;
            bL.x = Wl[(kb    ) * N_FEAT + n_off + l15];
            bL.y = Wl[(kb + 1) * N_FEAT + n_off + l15];
            bR.x = Wr[(kb    ) * N_FEAT + n_off + l15];
            bR.y = Wr[(kb + 1) * N_FEAT + n_off + l15];
            v2f aA0, aA1, aH0, aH1;
            aA0.x = lA[(l15     ) * N_FEAT + kb];
            aA0.y = lA[(l15     ) * N_FEAT + kb + 1];
            aA1.x = lA[(l15 + 16) * N_FEAT + kb];
            aA1.y = lA[(l15 + 16) * N_FEAT + kb + 1];
            aH0.x = lH[(l15     ) * N_FEAT + kb];
            aH0.y = lH[(l15     ) * N_FEAT + kb + 1];
            aH1.x = lH[(l15 + 16) * N_FEAT + kb];
            aH1.y = lH[(l15 + 16) * N_FEAT + kb + 1];
            cA0 = __builtin_amdgcn_wmma_f32_16x16x4_f32(false, aA0, false, bL,
                                                        (short)0, cA0, false, false);
            cA1 = __builtin_amdgcn_wmma_f32_16x16x4_f32(false, aA1, false, bL,
                                                        (short)0, cA1, false, false);
            cH0 = __builtin_amdgcn_wmma_f32_16x16x4_f32(false, aH0, false, bR,
                                                        (short)0, cH0, false, false);
            cH1 = __builtin_amdgcn_wmma_f32_16x16x4_f32(false, aH1, false, bR,
                                                        (short)0, cH1, false, false);
        }

        // ---- epilogue: per-row mean scale + bias(term) + ReLU + store ----
        const int rbase = (tile0 + t) * TILE_ROWS;
        float* outp = Out + (long long)rbase * N_FEAT + n_off + l15;
        if (rbase + TILE_ROWS <= n_rows) {
#pragma unroll
            for (int j = 0; j < 8; ++j) {
                int r0 = j + hi * 8;
                float v0 = fmaf(invdeg[rbase + r0],      cA0[j], cH0[j]);
                float v1 = fmaf(invdeg[rbase + r0 + 16], cA1[j], cH1[j]);
                v0 = v0 > 0.0f ? v0 : 0.0f;
                v1 = v1 > 0.0f ? v1 : 0.0f;
                outp[(long long)(r0     ) * N_FEAT] = v0;
                outp[(long long)(r0 + 16) * N_FEAT] = v1;
            }
        } else {
#pragma unroll
            for (int j = 0; j < 8; ++j) {
                int r0 = j + hi * 8;
                if (rbase + r0 < n_rows) {
                    float v = fmaf(invdeg[rbase + r0], cA0[j], cH0[j]);
                    outp[(long long)r0 * N_FEAT] = v > 0.0f ? v : 0.0f;
                }
                if (rbase + r0 + 16 < n_rows) {
                    float v = fmaf(invdeg[rbase + r0 + 16], cA1[j], cH1[j]);
                    outp[(long long)(r0 + 16) * N_FEAT] = v > 0.0f ? v : 0.0f;
                }
            }
        }
        __syncthreads();   // all reads of this buffer done before next prefetch
    }
}

// ------------------------------------------------- final 128->64 linear head
__global__ __launch_bounds__(128)
void gsage_final_linear(const float* __restrict__ H,
                        const float* __restrict__ W,    // [128,64]
                        const float* __restrict__ bias, // [64]
                        float* __restrict__ Out,
                        int n_rows) {
    __shared__ float ldsH[TILE_ELEMS];
    const int row0 = blockIdx.x * TILE_ROWS;
    const int tid  = threadIdx.x;

    for (int i = tid; i < TILE_ELEMS / 4; i += 128) {
        int r = i >> 5;
        int c = (i & 31) << 2;
        int grow = row0 + r;
        if (grow > n_rows - 1) grow = n_rows - 1;
        *(float4*)(ldsH + r * N_FEAT + c) =
            *(const float4*)(H + (long long)grow * N_FEAT + c);
    }
    __syncthreads();

    const int wave  = tid >> 5;     // 0..3 -> 64 cols
    const int lane  = tid & 31;
    const int n_off = wave * 16;
    const int l15   = lane & 15;
    const int hi    = lane >> 4;

    float bval = bias[n_off + l15];
    v8f c0, c1;
#pragma unroll
    for (int j = 0; j < 8; ++j) { c0[j] = bval; c1[j] = bval; }

#pragma unroll 4
    for (int kk = 0; kk < N_FEAT / 4; ++kk) {
        int kb = 4 * kk + (hi << 1);
        v2f b;
        b.x = W[(kb    ) * OUT_FEAT + n_off + l15];
        b.y = W[(kb + 1) * OUT_FEAT + n_off + l15];
        v2f a0, a1;
        a0.x = ldsH[(l15     ) * N_FEAT + kb];
        a0.y = ldsH[(l15     ) * N_FEAT + kb + 1];
        a1.x = ldsH[(l15 + 16) * N_FEAT + kb];
        a1.y = ldsH[(l15 + 16) * N_FEAT + kb + 1];
        c0 = __builtin_amdgcn_wmma_f32_16x16x4_f32(false, a0, false, b,
                                                   (short)0, c0, false, false);
        c1 = __builtin_amdgcn_wmma_f32_16x16x4_f32(false, a1, false, b,
                                                   (short)0, c1, false, false);
    }

    float* outp = Out + (long long)row0 * OUT_FEAT + n_off + l15;
    if (row0 + TILE_ROWS <= n_rows) {
#pragma unroll
        for (int j = 0; j < 8; ++j) {
            int r0 = j + hi * 8;
            outp[(long long)(r0     ) * OUT_FEAT] = c0[j];
            outp[(long long)(r0 + 16) * OUT_FEAT] = c1[j];
        }
    } else {
#pragma unroll
        for (int j = 0; j < 8; ++j) {
            int r0 = row0 + j + hi * 8;
            if (r0 < n_rows)      outp[(long long)(j + hi * 8     ) * OUT_FEAT] = c0[j];
            if (r0 + 16 < n_rows) outp[(long long)(j + hi * 8 + 16) * OUT_FEAT] = c1[j];
        }
    }
}

// ---------------------------------------------------------------- launcher
extern "C" void kernel_launch(void* const* d_in, const int* in_sizes, int n_in,
                              void* d_out, int out_size, void* d_ws, size_t ws_size,
                              hipStream_t stream) {
    const float*     x    = (const float*)d_in[0];
    const long long* edge = (const long long*)d_in[1];   // int64 per reference
    const float*     W1l  = (const float*)d_in[2];
    const float*     W1r  = (const float*)d_in[3];
    const float*     b1   = (const float*)d_in[4];
    const float*     W2l  = (const float*)d_in[5];
    const float*     W2r  = (const float*)d_in[6];
    const float*     b2   = (const float*)d_in[7];
    const float*     Wlin = (const float*)d_in[8];
    const float*     blin = (const float*)d_in[9];
    float*           out  = (float*)d_out;

    const int n_nodes = in_sizes[0] / N_FEAT;    // 100000
    const int n_edges = in_sizes[1] / 2;         // 1600000
    const long long* src = edge;
    const long long* dst = edge + n_edges;

    // workspace (~103 MB): deg | agg | h1 ; layer-2 output aliases agg (safe:
    // each tile's rows fully read before written; rows are block-private)
    char*  ws  = (char*)d_ws;
    size_t off = 0;
    float* deg = (float*)(ws + off); off += ((size_t)n_nodes * 4 + 255) & ~(size_t)255;
    float* agg = (float*)(ws + off); off += (size_t)n_nodes * N_FEAT * 4;
    float* h1  = (float*)(ws + off); off += (size_t)n_nodes * N_FEAT * 4;
    float* h2  = agg;
    (void)ws_size; (void)n_in; (void)out_size;

    const long long big = (long long)n_nodes * N_FEAT;
    const int scatter_blocks = (int)(((long long)n_edges * 32 + 255) / 256);
    const int n_tiles     = (n_nodes + TILE_ROWS - 1) / TILE_ROWS;
    const int sage_blocks = (n_tiles + TILES_PER_BLOCK - 1) / TILES_PER_BLOCK;
    const int head_blocks = n_tiles;

    // degrees -> reciprocal mean scale
    gsage_zero_f32<<<(n_nodes + 255) / 256, 256, 0, stream>>>(deg, n_nodes);
    gsage_degree<<<(n_edges + 255) / 256, 256, 0, stream>>>(dst, deg, n_edges);
    gsage_invdeg<<<(n_nodes + 255) / 256, 256, 0, stream>>>(deg, n_nodes);

    // layer 1
    gsage_zero_f32<<<4096, 256, 0, stream>>>(agg, big);
    gsage_scatter_add<<<scatter_blocks, 256, 0, stream>>>(x, src, dst, agg, n_edges);
    gsage_sage_layer<<<sage_blocks, 256, 0, stream>>>(agg, x, deg, W1l, W1r, b1,
                                                      h1, n_nodes, n_tiles);
    // layer 2 (output written in-place over agg)
    gsage_zero_f32<<<4096, 256, 0, stream>>>(agg, big);
    gsage_scatter_add<<<scatter_blocks, 256, 0, stream>>>(h1, src, dst, agg, n_edges);
    gsage_sage_layer<<<sage_blocks, 256, 0, stream>>>(agg, h1, deg, W2l, W2r, b2,
                                                      h2, n_nodes, n_tiles);
    // head
    gsage_final_linear<<<head_blocks, 128, 0, stream>>>(h2, Wlin, blin, out, n_nodes);
}